// AttentionBlock_6622839571024
// MI455X (gfx1250) — compile-verified
//
#include <hip/hip_runtime.h>
#include <hip/hip_bf16.h>

// CDNA5 fp32 WMMA: D(16x16) = A(16x4) * B(4x16) + C
typedef __attribute__((ext_vector_type(2))) float v2f;
typedef __attribute__((ext_vector_type(8))) float v8f;

__device__ __forceinline__ v8f wmma4(v2f a, v2f b, v8f c) {
  return __builtin_amdgcn_wmma_f32_16x16x4_f32(false, a, false, b, (short)0, c, false, false);
}

// Async DMA global->LDS, 16B per lane, tracked by ASYNCcnt (no VGPR staging).
__device__ __forceinline__ void async_copy_b128(uint32_t lds_byte_off, const float* gptr) {
  asm volatile("global_load_async_to_lds_b128 %0, %1, off"
               :: "v"(lds_byte_off), "v"(gptr) : "memory");
}
__device__ __forceinline__ void wait_async_le8() {
  asm volatile("s_wait_asynccnt 0x8" ::: "memory");
}
__device__ __forceinline__ void wait_async_0() {
  asm volatile("s_wait_asynccnt 0x0" ::: "memory");
}

#define LDST 68  // LDS row stride (floats): rows stay 16B-aligned (68*4=272=16*17)

// ---------------------------------------------------------------------------
// Kernel 1: QKV projection.  qkv[b,s,:] = xs[b,s,:] @ w_qkv + b_qkv
//   xs[b,s,c] = x[b,c,s]  (x is [B,C,S] with S=1024)
//   output column n (0..767): head h = n/192, segment -> Q/K/V buffers
//   Q/K/V stored as [b*4+h][s][64]
// Grid: (S/64=16, 768/64=12, B=16), block 256 (8 waves).
// Double-buffered LDS tiles filled by GLOBAL_LOAD_ASYNC_TO_LDS_B128.
// ---------------------------------------------------------------------------
__global__ __launch_bounds__(256) void qkv_gemm(
    const float* __restrict__ x, const float* __restrict__ w,
    const float* __restrict__ bias,
    float* __restrict__ Qb, float* __restrict__ Kb, float* __restrict__ Vb) {
  __shared__ float As[2][64 * LDST];  // As[k][m]   (x is k-major, m contiguous)
  __shared__ float Bs[2][64 * LDST];  // Bs[k][n]
  const int b = blockIdx.z, s0 = blockIdx.x * 64, n0 = blockIdx.y * 64;
  const int t = threadIdx.x;
  const int wave = t >> 5, lane = t & 31, lo = lane & 15, g = lane >> 4;
  const int mw = (wave & 3) * 16;   // wave's m offset within 64x64 tile
  const int nw = (wave >> 2) * 32;  // wave's n offset (covers nw, nw+16)
  const float* xb = x + (size_t)b * 256 * 1024;
  v8f c0 = {}; v8f c1 = {};

  const int f4 = (t & 15) * 4;
  const uint32_t ldsA0 = (uint32_t)(uintptr_t)(&As[0][0]);
  const uint32_t ldsB0 = (uint32_t)(uintptr_t)(&Bs[0][0]);
  const uint32_t bufStride = (uint32_t)(64 * LDST * 4);

  auto issue_chunk = [&](int k0, int bufi) {
    const uint32_t la0 = ldsA0 + bufi * bufStride;
    const uint32_t lb0 = ldsB0 + bufi * bufStride;
    int row = t >> 4;
    for (int p = 0; p < 4; ++p, row += 16) {
      const uint32_t loff = (uint32_t)(row * LDST + f4) * 4u;
      async_copy_b128(la0 + loff, xb + (size_t)(k0 + row) * 1024 + s0 + f4);
      async_copy_b128(lb0 + loff, w + (size_t)(k0 + row) * 768 + n0 + f4);
    }
  };

  issue_chunk(0, 0);
  for (int c = 0; c < 4; ++c) {
    if (c < 3) { issue_chunk(64 * (c + 1), (c + 1) & 1); wait_async_le8(); }
    else       { wait_async_0(); }
    __syncthreads();
    const float* Asb = As[c & 1];
    const float* Bsb = Bs[c & 1];
    for (int kb = 0; kb < 16; ++kb) {
      const int ka = 4 * kb + 2 * g;  // K index per A/B fragment layout
      v2f a;  a.x  = Asb[ka * LDST + mw + lo];       a.y  = Asb[(ka + 1) * LDST + mw + lo];
      v2f b0; b0.x = Bsb[ka * LDST + nw + lo];       b0.y = Bsb[(ka + 1) * LDST + nw + lo];
      v2f b1; b1.x = Bsb[ka * LDST + nw + 16 + lo];  b1.y = Bsb[(ka + 1) * LDST + nw + 16 + lo];
      c0 = wmma4(a, b0, c0);
      c1 = wmma4(a, b1, c1);
    }
    __syncthreads();
  }

  // Epilogue: this 64-wide column block maps wholly to Q, K, or V of head h.
  const int h = n0 / 192;
  const int seg = (n0 % 192) / 64;  // 0=Q, 1=K, 2=V
  float* dst = (seg == 0 ? Qb : (seg == 1 ? Kb : Vb)) + (size_t)(b * 4 + h) * 1024 * 64;
  for (int ct = 0; ct < 2; ++ct) {
    const int dcol = nw + ct * 16 + lo;
    const float bv = bias[n0 + dcol];
    v8f c = ct ? c1 : c0;
    for (int i = 0; i < 8; ++i) {
      const int s = s0 + mw + 8 * g + i;  // C layout: row = vgpr + 8*(lane/16)
      dst[(size_t)s * 64 + dcol] = c[i] + bv;
    }
  }
}

// ---------------------------------------------------------------------------
// Kernel 2: flash attention per (b,h). Each wave owns 16 queries; streaming
// softmax over 64 key blocks of 16. O accumulated as 4 fp32 C-fragments.
// Grid: (S/128=8, H=4, B=16), block 256 (8 independent waves).
// ---------------------------------------------------------------------------
__global__ __launch_bounds__(256) void attn_kernel(
    const float* __restrict__ Qb, const float* __restrict__ Kb,
    const float* __restrict__ Vb, float* __restrict__ Ob) {
  __shared__ float Pb[8][16 * 17];  // per-wave P-transpose staging
  const int wave = threadIdx.x >> 5, lane = threadIdx.x & 31;
  const int lo = lane & 15, g = lane >> 4;
  const int b = blockIdx.z, h = blockIdx.y;
  const int q0 = blockIdx.x * 128 + wave * 16;
  const size_t bh = (size_t)(b * 4 + h) * 1024 * 64;
  const float* Q = Qb + bh;
  const float* K = Kb + bh;
  const float* V = Vb + bh;

  // Preload Q A-fragments for all of d=64 (16 chunks of K=4).
  v2f qf[16];
  for (int kb = 0; kb < 16; ++kb) {
    const int d = 4 * kb + 2 * g;
    qf[kb] = *reinterpret_cast<const v2f*>(Q + (size_t)(q0 + lo) * 64 + d);
  }

  v8f o0 = {}, o1 = {}, o2 = {}, o3 = {};
  float m_i[8], l_i[8];
  for (int i = 0; i < 8; ++i) { m_i[i] = -1e30f; l_i[i] = 0.f; }
  float* P = Pb[wave];

  for (int j0 = 0; j0 < 1024; j0 += 16) {
    // ---- scores: S = Q @ K^T  (16x16 tile), inner dim 64
    v8f sc = {};
    for (int kb = 0; kb < 16; ++kb) {
      const int d = 4 * kb + 2 * g;
      v2f kf = *reinterpret_cast<const v2f*>(K + (size_t)(j0 + lo) * 64 + d);
      sc = wmma4(qf[kb], kf, sc);
    }
    // ---- online softmax (row = vgpr + 8*g, col = lo; reduce across 16 lanes)
    float p[8], rm[8];
    for (int i = 0; i < 8; ++i) { p[i] = sc[i] * 0.125f; rm[i] = p[i]; }
    for (int off = 1; off < 16; off <<= 1)
      for (int i = 0; i < 8; ++i)
        rm[i] = fmaxf(rm[i], __shfl_xor(rm[i], off, 32));
    float alpha[8];
    for (int i = 0; i < 8; ++i) {
      const float mn = fmaxf(m_i[i], rm[i]);
      alpha[i] = __expf(m_i[i] - mn);
      m_i[i] = mn;
      p[i] = __expf(p[i] - mn);
    }
    float rs[8];
    for (int i = 0; i < 8; ++i) rs[i] = p[i];
    for (int off = 1; off < 16; off <<= 1)
      for (int i = 0; i < 8; ++i) rs[i] += __shfl_xor(rs[i], off, 32);
    for (int i = 0; i < 8; ++i) l_i[i] = l_i[i] * alpha[i] + rs[i];
    for (int i = 0; i < 8; ++i) {
      o0[i] *= alpha[i]; o1[i] *= alpha[i]; o2[i] *= alpha[i]; o3[i] *= alpha[i];
    }
    // ---- transpose P (C-layout -> A-layout) through per-wave LDS
    for (int i = 0; i < 8; ++i) P[(i + 8 * g) * 17 + lo] = p[i];
    asm volatile("s_wait_dscnt 0" ::: "memory");
    // ---- O += P @ V  (16x16 x 16x64), 4 K-chunks x 4 N-blocks
    for (int kb = 0; kb < 4; ++kb) {
      const int kk = 4 * kb + 2 * g;
      v2f pa; pa.x = P[lo * 17 + kk]; pa.y = P[lo * 17 + kk + 1];
      const float* Vr0 = V + (size_t)(j0 + kk) * 64;
      const float* Vr1 = Vr0 + 64;
      v2f vf;
      vf.x = Vr0[lo];      vf.y = Vr1[lo];      o0 = wmma4(pa, vf, o0);
      vf.x = Vr0[16 + lo]; vf.y = Vr1[16 + lo]; o1 = wmma4(pa, vf, o1);
      vf.x = Vr0[32 + lo]; vf.y = Vr1[32 + lo]; o2 = wmma4(pa, vf, o2);
      vf.x = Vr0[48 + lo]; vf.y = Vr1[48 + lo]; o3 = wmma4(pa, vf, o3);
    }
  }

  for (int i = 0; i < 8; ++i) {
    const float inv = 1.f / l_i[i];
    o0[i] *= inv; o1[i] *= inv; o2[i] *= inv; o3[i] *= inv;
  }
  // Store attn output gathered as [b][s][h*64 + d]
  for (int i = 0; i < 8; ++i) {
    const int s = q0 + i + 8 * g;
    float* dst = Ob + ((size_t)(b * 1024 + s)) * 256 + h * 64;
    dst[lo]      = o0[i];
    dst[16 + lo] = o1[i];
    dst[32 + lo] = o2[i];
    dst[48 + lo] = o3[i];
  }
}

// ---------------------------------------------------------------------------
// Kernel 3: out = attnOut @ w_out + b_out + xs, transposed to [B,C,S].
// Grid: (S/64=16, 256/64=4, B=16), block 256. Same async double-buffering.
// ---------------------------------------------------------------------------
__global__ __launch_bounds__(256) void out_gemm(
    const float* __restrict__ A, const float* __restrict__ w,
    const float* __restrict__ bias, const float* __restrict__ x,
    float* __restrict__ out) {
  __shared__ float As[2][64 * LDST];  // As[m][k]  (A is m-major, k contiguous)
  __shared__ float Bs[2][64 * LDST];  // Bs[k][n]
  const int b = blockIdx.z, s0 = blockIdx.x * 64, n0 = blockIdx.y * 64;
  const int t = threadIdx.x;
  const int wave = t >> 5, lane = t & 31, lo = lane & 15, g = lane >> 4;
  const int mw = (wave & 3) * 16, nw = (wave >> 2) * 32;
  const float* Ab = A + (size_t)b * 1024 * 256;
  v8f c0 = {}; v8f c1 = {};

  const int f4 = (t & 15) * 4;
  const uint32_t ldsA0 = (uint32_t)(uintptr_t)(&As[0][0]);
  const uint32_t ldsB0 = (uint32_t)(uintptr_t)(&Bs[0][0]);
  const uint32_t bufStride = (uint32_t)(64 * LDST * 4);

  auto issue_chunk = [&](int k0, int bufi) {
    const uint32_t la0 = ldsA0 + bufi * bufStride;
    const uint32_t lb0 = ldsB0 + bufi * bufStride;
    int row = t >> 4;
    for (int p = 0; p < 4; ++p, row += 16) {
      const uint32_t loff = (uint32_t)(row * LDST + f4) * 4u;
      async_copy_b128(la0 + loff, Ab + (size_t)(s0 + row) * 256 + k0 + f4);  // As[m][k]
      async_copy_b128(lb0 + loff, w + (size_t)(k0 + row) * 256 + n0 + f4);   // Bs[k][n]
    }
  };

  issue_chunk(0, 0);
  for (int c = 0; c < 4; ++c) {
    if (c < 3) { issue_chunk(64 * (c + 1), (c + 1) & 1); wait_async_le8(); }
    else       { wait_async_0(); }
    __syncthreads();
    const float* Asb = As[c & 1];
    const float* Bsb = Bs[c & 1];
    for (int kb = 0; kb < 16; ++kb) {
      const int ka = 4 * kb + 2 * g;
      v2f a;  a.x  = Asb[(mw + lo) * LDST + ka];     a.y  = Asb[(mw + lo) * LDST + ka + 1];
      v2f b0; b0.x = Bsb[ka * LDST + nw + lo];       b0.y = Bsb[(ka + 1) * LDST + nw + lo];
      v2f b1; b1.x = Bsb[ka * LDST + nw + 16 + lo];  b1.y = Bsb[(ka + 1) * LDST + nw + 16 + lo];
      c0 = wmma4(a, b0, c0);
      c1 = wmma4(a, b1, c1);
    }
    __syncthreads();
  }

  const float* xb = x + (size_t)b * 256 * 1024;
  float* ob = out + (size_t)b * 256 * 1024;
  for (int ct = 0; ct < 2; ++ct) {
    const int c = n0 + nw + ct * 16 + lo;  // output channel
    const float bv = bias[c];
    v8f cc = ct ? c1 : c0;
    for (int i = 0; i < 8; ++i) {
      const int s = s0 + mw + 8 * g + i;
      const size_t idx = (size_t)c * 1024 + s;
      ob[idx] = cc[i] + bv + xb[idx];  // + residual xs[b,s,c] = x[b,c,s]
    }
  }
}

// ---------------------------------------------------------------------------
extern "C" void kernel_launch(void* const* d_in, const int* in_sizes, int n_in,
                              void* d_out, int out_size, void* d_ws, size_t ws_size,
                              hipStream_t stream) {
  const float* x     = (const float*)d_in[0];
  const float* w_qkv = (const float*)d_in[1];
  const float* b_qkv = (const float*)d_in[2];
  const float* w_out = (const float*)d_in[3];
  const float* b_out = (const float*)d_in[4];

  float* ws = (float*)d_ws;
  // Layout (floats): Q | K | V | attnOut, each 16*4*1024*64 = 4194304 (16 MB)
  float* Qb = ws;
  float* Kb = ws + 4194304u;
  float* Vb = ws + 8388608u;
  float* Ob = ws + 12582912u;  // [B,S,256] = 4194304 floats

  qkv_gemm<<<dim3(16, 12, 16), 256, 0, stream>>>(x, w_qkv, b_qkv, Qb, Kb, Vb);
  attn_kernel<<<dim3(8, 4, 16), 256, 0, stream>>>(Qb, Kb, Vb, Ob);
  out_gemm<<<dim3(16, 4, 16), 256, 0, stream>>>(Ob, w_out, b_out, x, (float*)d_out);
}